// TemporalSelfAttention_25821343384036
// MI455X (gfx1250) — compile-verified
//
#include <hip/hip_runtime.h>
#include <hip/hip_bf16.h>
#include <math.h>

// ---------------------------------------------------------------------------
// Temporal (deformable) self-attention for MI455X / gfx1250.
//   nq = H*W queries, EMBED=256, 8 heads x 32 dim, 2 queues, 1 level, 4 points
// Pipeline (4 kernels, all f32, GEMMs on V_WMMA_F32_16X16X4_F32):
//   1) val   = query @ W_v  + b_v                      (nq x 256)
//   2) off   = qcat @ W_off + b_off   (nq x 128)
//      attn  = qcat @ W_attn+ b_attn  (nq x 64)   [qcat synthesized on load]
//   3) smean = queue-mean of softmax-weighted bilinear samples (nq x 256)
//   4) out   = smean @ W_out + b_out + query           (nq x 256)
// B (weight) slabs are staged to LDS once per block via ASYNC load-to-LDS
// (transposed, +4 pad so ds_load_b64 pairs are bank-conflict free), shared
// by all 8 waves.
// ---------------------------------------------------------------------------

typedef float v2f __attribute__((ext_vector_type(2)));
typedef float v8f __attribute__((ext_vector_type(8)));

#define EMBED      256
#define NUM_HEADS  8
#define HEAD_DIM   32
#define NUM_QUEUE  2
#define NUM_POINTS 4

#ifndef __has_builtin
#define __has_builtin(x) 0
#endif

#if __has_builtin(__builtin_amdgcn_global_load_async_to_lds_b32) && \
    __has_builtin(__builtin_amdgcn_s_wait_asynccnt)
#define USE_ASYNC_LDS 1
typedef __attribute__((address_space(1))) int g_i32_t;   // global int*
typedef __attribute__((address_space(3))) int l_i32_t;   // LDS int*
#else
#define USE_ASYNC_LDS 0
#endif

__device__ __forceinline__ v8f wmma_k4(v2f a, v2f b, v8f c) {
  // D = A(16x4,f32) * B(4x16,f32) + C(16x16,f32)
  return __builtin_amdgcn_wmma_f32_16x16x4_f32(
      /*neg_a=*/false, a, /*neg_b=*/false, b,
      /*c_mod=*/(short)0, c, /*reuse_a=*/false, /*reuse_b=*/false);
}

// Stage B[k][n0+c] (k=0..K-1, c=0..15) into LDS transposed: lb[c*ldk + k].
// ldk = K+4 -> ds_load_b64 of (k0,k0+1) at bank (4r + k0) % 64: conflict-free.
__device__ __forceinline__
void stage_B(const float* __restrict__ B, float* lb, int K, int N, int n0, int ldk) {
#if USE_ASYNC_LDS
  for (int idx = threadIdx.x; idx < K * 16; idx += 256) {
    const int k = idx >> 4, c = idx & 15;
    __builtin_amdgcn_global_load_async_to_lds_b32(
        (g_i32_t*)(B + (size_t)k * N + n0 + c),
        (l_i32_t*)(lb + c * ldk + k), 0, 0);
  }
  __builtin_amdgcn_s_wait_asynccnt(0);
#else
  for (int idx = threadIdx.x; idx < K * 16; idx += 256) {
    const int k = idx >> 4, c = idx & 15;
    lb[c * ldk + k] = B[(size_t)k * N + n0 + c];
  }
#endif
  __syncthreads();
}

// ---------------------------------------------------------------------------
// GEMM: C[M x N=256] = A[M x 256] * B[256 x 256] + bias (+ optional ident)
// One wave per 16x16 output tile; 8 waves / block; grid.y = N/16 tiles.
// ---------------------------------------------------------------------------
template <bool ADD_IDENT>
__device__ __forceinline__
void gemm256_body(const float* __restrict__ A, const float* __restrict__ B,
                  const float* __restrict__ bias, const float* __restrict__ ident,
                  float* __restrict__ C, int M) {
  __shared__ float lb[16 * (256 + 4)];
  const int ldk = 256 + 4;
  const int n0  = blockIdx.y << 4;

  stage_B(B, lb, 256, 256, n0, ldk);                 // all threads participate

  const int lane = threadIdx.x & 31;
  const int half = lane >> 4;          // 0: K pair {0,1}, 1: K pair {2,3}
  const int r    = lane & 15;
  const int mt   = blockIdx.x * 8 + (threadIdx.x >> 5);
  const int mtiles = (M + 15) >> 4;
  if (mt >= mtiles) return;            // after barrier; wave-uniform
  const int m0 = mt << 4;

  int arow = m0 + r; if (arow >= M) arow = M - 1;    // clamp tail rows
  const float* __restrict__ Ar = A + (size_t)arow * EMBED;
  const float* __restrict__ Br = lb + r * ldk;       // this lane's B column

  v8f c = {};
#pragma unroll 8
  for (int kb = 0; kb < 256; kb += 4) {
    const int k0 = kb + 2 * half;
    v2f a; a.x = Ar[k0]; a.y = Ar[k0 + 1];
    v2f b; b.x = Br[k0]; b.y = Br[k0 + 1];
    c = wmma_k4(a, b, c);
  }

  const float bn = bias[n0 + r];
  if (m0 + 16 <= M) {                                // full tile: no guards
#pragma unroll
    for (int i = 0; i < 8; ++i) {
      const size_t o = (size_t)(m0 + i + 8 * half) * 256 + n0 + r;
      C[o] = ADD_IDENT ? (c[i] + bn + ident[o]) : (c[i] + bn);
    }
  } else {
#pragma unroll
    for (int i = 0; i < 8; ++i) {
      const int m = m0 + i + 8 * half;
      if (m < M) {
        const size_t o = (size_t)m * 256 + n0 + r;
        C[o] = ADD_IDENT ? (c[i] + bn + ident[o]) : (c[i] + bn);
      }
    }
  }
}

__global__ __launch_bounds__(256)
void k_gemm_val(const float* __restrict__ A, const float* __restrict__ B,
                const float* __restrict__ bias, float* __restrict__ C, int M) {
  gemm256_body<false>(A, B, bias, nullptr, C, M);
}

__global__ __launch_bounds__(256)
void k_gemm_out(const float* __restrict__ A, const float* __restrict__ B,
                const float* __restrict__ bias, const float* __restrict__ ident,
                float* __restrict__ O, int M) {
  gemm256_body<true>(A, B, bias, ident, O, M);
}

// ---------------------------------------------------------------------------
// GEMM over implicit qcat = [query | query+query_pos]  (K = 512).
// grid.y in [0,12): tiles 0..7 -> W_off (N=128), tiles 8..11 -> W_attn (N=64).
// ---------------------------------------------------------------------------
__global__ __launch_bounds__(256)
void k_gemm_offattn(const float* __restrict__ query, const float* __restrict__ qpos,
                    const float* __restrict__ Woff,  const float* __restrict__ boff,
                    const float* __restrict__ Wattn, const float* __restrict__ battn,
                    float* __restrict__ off, float* __restrict__ attn, int M) {
  __shared__ float lb[16 * (512 + 4)];
  const int ldk = 512 + 4;

  const bool  isAttn = blockIdx.y >= 8;
  const int   N      = isAttn ? 64 : 128;
  const int   n0     = (isAttn ? ((int)blockIdx.y - 8) : (int)blockIdx.y) << 4;
  const float* __restrict__ B    = isAttn ? Wattn : Woff;
  const float* __restrict__ bias = isAttn ? battn : boff;
  float* __restrict__ Out        = isAttn ? attn  : off;

  stage_B(B, lb, 512, N, n0, ldk);                   // all threads participate

  const int lane = threadIdx.x & 31;
  const int half = lane >> 4;
  const int r    = lane & 15;
  const int mt   = blockIdx.x * 8 + (threadIdx.x >> 5);
  const int mtiles = (M + 15) >> 4;
  if (mt >= mtiles) return;
  const int m0 = mt << 4;

  int arow = m0 + r; if (arow >= M) arow = M - 1;
  const float* __restrict__ Aq = query + (size_t)arow * EMBED;
  const float* __restrict__ Ap = qpos  + (size_t)arow * EMBED;
  const float* __restrict__ Br = lb + r * ldk;

  v8f c = {};
  // K = 0..255 : A = query, W rows 0..255
#pragma unroll 8
  for (int kb = 0; kb < 256; kb += 4) {
    const int k0 = kb + 2 * half;
    v2f a; a.x = Aq[k0]; a.y = Aq[k0 + 1];
    v2f b; b.x = Br[k0]; b.y = Br[k0 + 1];
    c = wmma_k4(a, b, c);
  }
  // K = 256..511 : A = query + query_pos, W rows 256..511
#pragma unroll 8
  for (int kb = 0; kb < 256; kb += 4) {
    const int k0 = kb + 2 * half;
    v2f a; a.x = Aq[k0] + Ap[k0]; a.y = Aq[k0 + 1] + Ap[k0 + 1];
    v2f b; b.x = Br[256 + k0];    b.y = Br[257 + k0];
    c = wmma_k4(a, b, c);
  }

  const float bn = bias[n0 + r];
  if (m0 + 16 <= M) {
#pragma unroll
    for (int i = 0; i < 8; ++i)
      Out[(size_t)(m0 + i + 8 * half) * N + n0 + r] = c[i] + bn;
  } else {
#pragma unroll
    for (int i = 0; i < 8; ++i) {
      const int m = m0 + i + 8 * half;
      if (m < M) Out[(size_t)m * N + n0 + r] = c[i] + bn;
    }
  }
}

// ---------------------------------------------------------------------------
// Softmax(4 pts) + bilinear sampling + queue-mean.
// One wave per (query, head); lane == channel (HEAD_DIM == wave32).
// off layout:  q*128 + h*16 + qu*8 + p*2 + {x,y}
// attn layout: q*64  + h*8  + qu*4 + p
// ---------------------------------------------------------------------------
__global__ __launch_bounds__(256)
void k_sample(const float* __restrict__ val, const float* __restrict__ offb,
              const float* __restrict__ attnb, const float* __restrict__ ref,
              const int* __restrict__ ss, float* __restrict__ smean, int M) {
  const int w = blockIdx.x * 8 + (threadIdx.x >> 5);
  const int q = w >> 3;
  const int h = w & 7;
  if (q >= M) return;                  // wave-uniform
  const int d = threadIdx.x & 31;

  const int   Hs = ss[0], Ws = ss[1];
  const float fW = (float)Ws, fH = (float)Hs;
  const float* __restrict__ vh = val + h * HEAD_DIM + d;  // stride 256 / pixel

  float acc = 0.f;
#pragma unroll
  for (int qu = 0; qu < NUM_QUEUE; ++qu) {
    const float rx = ref[((size_t)qu * M + q) * 2 + 0];
    const float ry = ref[((size_t)qu * M + q) * 2 + 1];

    const float* lg = attnb + (size_t)q * 64 + h * 8 + qu * 4;
    const float l0 = lg[0], l1 = lg[1], l2 = lg[2], l3 = lg[3];
    const float mx = fmaxf(fmaxf(l0, l1), fmaxf(l2, l3));
    const float e0 = __expf(l0 - mx), e1 = __expf(l1 - mx);
    const float e2 = __expf(l2 - mx), e3 = __expf(l3 - mx);
    const float inv = 1.f / (e0 + e1 + e2 + e3);
    const float wt[4] = { e0 * inv, e1 * inv, e2 * inv, e3 * inv };

    const float* of = offb + (size_t)q * 128 + h * 16 + qu * 8;
#pragma unroll
    for (int p = 0; p < NUM_POINTS; ++p) {
      // loc*W - 0.5 = ref*W + off - 0.5  (normalizer W cancels)
      const float x = rx * fW + of[p * 2 + 0] - 0.5f;
      const float y = ry * fH + of[p * 2 + 1] - 0.5f;
      const float xf = floorf(x), yf = floorf(y);
      const float wx = x - xf,  wy = y - yf;
      const int   xi = (int)xf, yi = (int)yf;

      float s = 0.f;
      if (xi >= 0 && xi < Ws && yi >= 0 && yi < Hs)
        s += vh[((size_t)yi * Ws + xi) * EMBED] * (1.f - wx) * (1.f - wy);
      if (xi + 1 >= 0 && xi + 1 < Ws && yi >= 0 && yi < Hs)
        s += vh[((size_t)yi * Ws + xi + 1) * EMBED] * wx * (1.f - wy);
      if (xi >= 0 && xi < Ws && yi + 1 >= 0 && yi + 1 < Hs)
        s += vh[((size_t)(yi + 1) * Ws + xi) * EMBED] * (1.f - wx) * wy;
      if (xi + 1 >= 0 && xi + 1 < Ws && yi + 1 >= 0 && yi + 1 < Hs)
        s += vh[((size_t)(yi + 1) * Ws + xi + 1) * EMBED] * wx * wy;

      acc += wt[p] * s;
    }
  }
  smean[(size_t)q * EMBED + h * HEAD_DIM + d] = 0.5f * acc;  // mean over 2 queues
}

// ---------------------------------------------------------------------------
extern "C" void kernel_launch(void* const* d_in, const int* in_sizes, int n_in,
                              void* d_out, int out_size, void* d_ws, size_t ws_size,
                              hipStream_t stream) {
  const float* query  = (const float*)d_in[0];
  const float* qpos   = (const float*)d_in[1];
  const float* ref    = (const float*)d_in[2];
  const float* W_off  = (const float*)d_in[3];
  const float* b_off  = (const float*)d_in[4];
  const float* W_attn = (const float*)d_in[5];
  const float* b_attn = (const float*)d_in[6];
  const float* W_v    = (const float*)d_in[7];
  const float* b_v    = (const float*)d_in[8];
  const float* W_out  = (const float*)d_in[9];
  const float* b_out  = (const float*)d_in[10];
  const int*   ss     = (const int*)d_in[11];

  const int M = in_sizes[0] / EMBED;   // nq = H*W

  float* ws    = (float*)d_ws;
  float* val   = ws;                          // M x 256
  float* off   = val  + (size_t)M * 256;      // M x 128
  float* attn  = off  + (size_t)M * 128;      // M x 64
  float* smean = attn + (size_t)M * 64;       // M x 256

  const int mtiles  = (M + 15) / 16;
  const int mblocks = (mtiles + 7) / 8;       // 8 waves (tiles) per block
  const dim3 blk(256);

  k_gemm_val    <<<dim3(mblocks, 16), blk, 0, stream>>>(query, W_v, b_v, val, M);
  k_gemm_offattn<<<dim3(mblocks, 12), blk, 0, stream>>>(query, qpos, W_off, b_off,
                                                        W_attn, b_attn, off, attn, M);
  const int swaves = M * NUM_HEADS;
  k_sample      <<<dim3((swaves + 7) / 8), blk, 0, stream>>>(val, off, attn, ref, ss,
                                                             smean, M);
  k_gemm_out    <<<dim3(mblocks, 16), blk, 0, stream>>>(smean, W_out, b_out, query,
                                                        (float*)d_out, M);
}